// MambaEncoder_25606595018911
// MI455X (gfx1250) — compile-verified
//
#include <hip/hip_runtime.h>
#include <hip/hip_bf16.h>
#include <math.h>

// ---------------- problem constants (match reference) ----------------
#define BB      2
#define LL      1024
#define DM      512          // d_model
#define EE      1024         // d_inner
#define NS      16           // d_state
#define RR      32           // dt_rank
#define DCONV   4
#define TOK     (BB * LL)    // 2048 rows
#define EPSF    1.1920929e-07f
#define ROWW    (RR + 2 * NS)   // 64 floats: [dt_raw(32) | B(16) | C(16)]

typedef __attribute__((ext_vector_type(2))) float v2f;
typedef __attribute__((ext_vector_type(8))) float v8f;
typedef __attribute__((ext_vector_type(4))) int   v4i;

#if defined(__gfx1250__) && __has_builtin(__builtin_amdgcn_global_load_async_to_lds_b128) && __has_builtin(__builtin_amdgcn_s_wait_asynccnt)
#define USE_ASYNC_LDS 1
#else
#define USE_ASYNC_LDS 0
#endif

// ---------------------------------------------------------------------
// GEMM: C[M,N] = X[M,K] (row-major) * W[N,K]^T (W row-major over K)
// One wave -> 32x64 macro-tile = 2x4 grid of 16x16 fp32 WMMA tiles.
// Software-pipelined: double-buffered operand registers; the next
// k-step's 6 loads issue before the current stage's 8 WMMAs consume,
// so load latency hides under the XDL pipe. K % 8 == 0 for all calls.
// ---------------------------------------------------------------------
#define MT 2   // m-tiles per wave (32 rows)
#define NT 4   // n-tiles per wave (64 cols)

__global__ __launch_bounds__(256) void wmma_gemm_xwt_kernel(
    const float* __restrict__ X, const float* __restrict__ W,
    float* __restrict__ C, int M, int N, int K)
{
    const int wave   = blockIdx.x * (blockDim.x >> 5) + (threadIdx.x >> 5);
    const int tilesN = N >> 6;                 // macro-cols of 64
    const int wm = wave / tilesN;              // macro-row (32 rows)
    const int wn = wave % tilesN;
    if (wm * (MT * 16) >= M) return;

    const int lane = threadIdx.x & 31;
    const int half = lane >> 4;                // 0: K+0..1 , 1: K+2..3
    const int l16  = lane & 15;

    const float* __restrict__ xbase = X + (size_t)(wm * (MT * 16) + l16) * K + 2 * half;
    const float* __restrict__ wbase = W + (size_t)(wn * (NT * 16) + l16) * K + 2 * half;

    v8f acc[MT][NT] = {};
    v2f a0[MT], b0[NT], a1[MT], b1[NT];

    auto loadAB = [&](v2f* a, v2f* b, int k) {
        #pragma unroll
        for (int i = 0; i < MT; ++i)
            a[i] = *(const v2f*)(xbase + (size_t)i * 16 * K + k);
        #pragma unroll
        for (int j = 0; j < NT; ++j)
            b[j] = *(const v2f*)(wbase + (size_t)j * 16 * K + k);
    };
    auto mmaAB = [&](const v2f* a, const v2f* b) {
        #pragma unroll
        for (int i = 0; i < MT; ++i)
            #pragma unroll
            for (int j = 0; j < NT; ++j)
                acc[i][j] = __builtin_amdgcn_wmma_f32_16x16x4_f32(
                    false, a[i], false, b[j], (short)0, acc[i][j], false, false);
    };

    loadAB(a0, b0, 0);                       // prologue
    for (int k = 0; k < K; k += 8) {
        __builtin_prefetch(xbase + k + 128, 0, 3);
        __builtin_prefetch(wbase + k + 128, 0, 3);
        loadAB(a1, b1, k + 4);               // k+4 < K since K % 8 == 0
        mmaAB(a0, b0);
        if (k + 8 < K) loadAB(a0, b0, k + 8);
        mmaAB(a1, b1);
    }

    // C/D layout: VGPR v -> row (v + 8*half), col l16
    #pragma unroll
    for (int i = 0; i < MT; ++i)
        #pragma unroll
        for (int j = 0; j < NT; ++j)
            #pragma unroll
            for (int v = 0; v < 8; ++v)
                C[(size_t)(wm * (MT * 16) + i * 16 + v + 8 * half) * N
                  + wn * (NT * 16) + j * 16 + l16] = acc[i][j][v];
}

// ---------------------------------------------------------------------
// h + pos_emb, RMSNorm -> xn   (one 512-wide row per block)
// ---------------------------------------------------------------------
__global__ __launch_bounds__(256) void add_rmsnorm_kernel(
    const float* __restrict__ h, const float* __restrict__ pos,
    const float* __restrict__ w, float* __restrict__ xn)
{
    __shared__ float red[256];
    const int row = blockIdx.x;
    const int t   = threadIdx.x;
    const float* hp = h   + (size_t)row * DM;
    const float* pp = pos + (size_t)row * DM;

    float t0 = hp[t]       + pp[t];
    float t1 = hp[t + 256] + pp[t + 256];
    red[t] = t0 * t0 + t1 * t1;
    __syncthreads();
    #pragma unroll
    for (int s = 128; s > 0; s >>= 1) {
        if (t < s) red[t] += red[t + s];
        __syncthreads();
    }
    const float rs = rsqrtf(red[0] * (1.0f / DM) + EPSF);
    float* o = xn + (size_t)row * DM;
    o[t]       = t0 * rs * w[t];
    o[t + 256] = t1 * rs * w[t + 256];
}

// ---------------------------------------------------------------------
// Causal depthwise conv1d (k=4) + bias + SiLU on xi = xz[:, :, :E]
// ---------------------------------------------------------------------
__global__ __launch_bounds__(256) void conv_silu_kernel(
    const float* __restrict__ xz, const float* __restrict__ cw,
    const float* __restrict__ cb, float* __restrict__ xi)
{
    const int idx = blockIdx.x * blockDim.x + threadIdx.x;   // b*L*E grid
    const int e = idx & (EE - 1);
    const int l = (idx >> 10) & (LL - 1);
    const int b = idx >> 20;

    float acc = cb[e];
    #pragma unroll
    for (int j = 0; j < DCONV; ++j) {
        const int ls = l - (DCONV - 1) + j;
        if (ls >= 0)
            acc += cw[e * DCONV + j] * xz[((size_t)(b * LL + ls)) * (2 * EE) + e];
    }
    xi[idx] = acc / (1.0f + __expf(-acc));   // SiLU
}

// ---------------------------------------------------------------------
// Fused: dt = softplus(dbc_dt @ dt_w^T + dt_b), selective scan, +u*D,
// gate with silu(z).  In-place: xi holds u on entry, gated y on exit.
// One thread per channel e; 16-state recurrence in registers.  The
// shared 64-float dbc row is DOUBLE-BUFFERED in LDS: the CDNA5 async
// global->LDS engine runs one timestep ahead (issue row l+1, then
// s_wait_asynccnt(1) guarantees row l landed since async loads
// complete in order), hiding load latency behind the step's VALU work.
// ---------------------------------------------------------------------
__global__ __launch_bounds__(256) void scan_gate_kernel(
    const float* __restrict__ dbc, const float* __restrict__ xz,
    const float* __restrict__ dtw, const float* __restrict__ dtb,
    const float* __restrict__ Alog, const float* __restrict__ Dv,
    float* __restrict__ xi)
{
    const int b = blockIdx.x / (EE / 256);
    const int e = (blockIdx.x % (EE / 256)) * 256 + threadIdx.x;

    float wdt[RR];
    #pragma unroll
    for (int r = 0; r < RR; ++r) wdt[r] = dtw[e * RR + r];
    float Arow[NS];
    #pragma unroll
    for (int n = 0; n < NS; ++n) Arow[n] = -__expf(Alog[e * NS + n]);
    float hst[NS];
    #pragma unroll
    for (int n = 0; n < NS; ++n) hst[n] = 0.0f;
    const float dcoef = Dv[e];
    const float bias  = dtb[e];

    __shared__ float rowbuf[2][ROWW];

#if USE_ASYNC_LDS
    auto issue_row = [&](int l) {
        if (threadIdx.x < 16) {
            const float* g = dbc + (size_t)(b * LL + l) * ROWW + (size_t)threadIdx.x * 4;
            __builtin_amdgcn_global_load_async_to_lds_b128(
                (__attribute__((address_space(1))) v4i*)(uintptr_t)g,
                (__attribute__((address_space(3))) v4i*)(uint32_t)(uintptr_t)&rowbuf[l & 1][threadIdx.x * 4],
                0, 0);
        }
    };
    issue_row(0);                      // prologue: row 0 in flight
#endif

    for (int l = 0; l < LL; ++l) {
        const size_t tok = (size_t)(b * LL + l);
        __syncthreads();               // iteration l-1 done reading rowbuf[(l+1)&1]
#if USE_ASYNC_LDS
        if (l + 1 < LL) {
            issue_row(l + 1);
            __builtin_amdgcn_s_wait_asynccnt(1);   // row l landed (in-order)
        } else {
            __builtin_amdgcn_s_wait_asynccnt(0);
        }
#else
        if (threadIdx.x < ROWW)
            rowbuf[l & 1][threadIdx.x] = dbc[tok * ROWW + threadIdx.x];
#endif
        __syncthreads();
        const float* row = rowbuf[l & 1];

        const float u = xi[tok * EE + e];
        const float z = xz[tok * (2 * EE) + EE + e];

        float dtr = bias;
        #pragma unroll
        for (int r = 0; r < RR; ++r) dtr += wdt[r] * row[r];
        const float dt = (dtr > 20.0f) ? dtr : log1pf(__expf(dtr));   // softplus

        const float du = dt * u;
        float y = 0.0f;
        #pragma unroll
        for (int n = 0; n < NS; ++n) {
            const float dA = __expf(dt * Arow[n]);
            hst[n] = dA * hst[n] + du * row[RR + n];
            y += hst[n] * row[RR + NS + n];
        }
        y += u * dcoef;
        const float sz = z / (1.0f + __expf(-z));
        xi[tok * EE + e] = y * sz;
    }
}

// ---------------------------------------------------------------------
// host-side launch
// ---------------------------------------------------------------------
extern "C" void kernel_launch(void* const* d_in, const int* in_sizes, int n_in,
                              void* d_out, int out_size, void* d_ws, size_t ws_size,
                              hipStream_t stream)
{
    const float* x        = (const float*)d_in[0];
    const float* pos      = (const float*)d_in[1];
    const float* norm_w   = (const float*)d_in[2];
    const float* in_proj  = (const float*)d_in[3];   // (6, 2E, DM)
    const float* conv_w   = (const float*)d_in[4];   // (6, E, 4)
    const float* conv_b   = (const float*)d_in[5];   // (6, E)
    const float* x_proj   = (const float*)d_in[6];   // (6, R+2N, E)
    const float* dt_w     = (const float*)d_in[7];   // (6, E, R)
    const float* dt_b     = (const float*)d_in[8];   // (6, E)
    const float* A_log    = (const float*)d_in[9];   // (6, E, N)
    const float* Dvec     = (const float*)d_in[10];  // (6, E)
    const float* out_w    = (const float*)d_in[11];  // (6, DM, E)
    float* out            = (float*)d_out;

    // workspace layout (floats)
    float* ws   = (float*)d_ws;
    float* hbuf = ws;                               // TOK*DM
    float* xn   = hbuf + (size_t)TOK * DM;          // TOK*DM
    float* xz   = xn   + (size_t)TOK * DM;          // TOK*2E
    float* xi   = xz   + (size_t)TOK * 2 * EE;      // TOK*E
    float* dbc  = xi   + (size_t)TOK * EE;          // TOK*ROWW

    const int NLAYERS = 6;
    for (int layer = 0; layer < NLAYERS; ++layer) {
        const float* hin  = (layer == 0) ? x : hbuf;
        float*       hout = (layer == NLAYERS - 1) ? out : hbuf;

        const float* ipw = in_proj + (size_t)layer * 2 * EE * DM;
        const float* cw  = conv_w  + (size_t)layer * EE * DCONV;
        const float* cb  = conv_b  + (size_t)layer * EE;
        const float* xpw = x_proj  + (size_t)layer * ROWW * EE;
        const float* dw  = dt_w    + (size_t)layer * EE * RR;
        const float* db  = dt_b    + (size_t)layer * EE;
        const float* Al  = A_log   + (size_t)layer * EE * NS;
        const float* Dl  = Dvec    + (size_t)layer * EE;
        const float* ow  = out_w   + (size_t)layer * DM * EE;

        // 1) h + pos, RMSNorm -> xn
        add_rmsnorm_kernel<<<TOK, 256, 0, stream>>>(hin, pos, norm_w, xn);

        // 2) xz = xn @ in_proj^T   (2048 x 2048, K=512)
        {
            int waves = (TOK / (MT * 16)) * (2 * EE / (NT * 16));
            wmma_gemm_xwt_kernel<<<(waves + 7) / 8, 256, 0, stream>>>(
                xn, ipw, xz, TOK, 2 * EE, DM);
        }

        // 3) depthwise causal conv + SiLU -> xi
        conv_silu_kernel<<<(TOK * EE) / 256, 256, 0, stream>>>(xz, cw, cb, xi);

        // 4) dbc = xi @ x_proj^T   (2048 x 64, K=1024)
        {
            int waves = (TOK / (MT * 16)) * (ROWW / (NT * 16));
            wmma_gemm_xwt_kernel<<<(waves + 7) / 8, 256, 0, stream>>>(
                xi, xpw, dbc, TOK, ROWW, EE);
        }

        // 5) fused dt-proj + softplus + selective scan + D skip + SiLU gate
        scan_gate_kernel<<<BB * (EE / 256), 256, 0, stream>>>(
            dbc, xz, dw, db, Al, Dl, xi);

        // 6) h_next = y @ out_w^T   (2048 x 512, K=1024)
        {
            int waves = (TOK / (MT * 16)) * (DM / (NT * 16));
            wmma_gemm_xwt_kernel<<<(waves + 7) / 8, 256, 0, stream>>>(
                xi, ow, hout, TOK, DM, EE);
        }
    }
}